// LFAggregationModule_48962627174704
// MI455X (gfx1250) — compile-verified
//
#include <hip/hip_runtime.h>

// ---------------------------------------------------------------------------
// Problem constants (from the reference)
// ---------------------------------------------------------------------------
#define B_      8
#define N_PER   8192
#define DEC     4
#define KNBR    16
#define F_IN    64
#define HID     64
#define F_OUT   128
#define M_PER   (N_PER / DEC)       // 2048 queries per batch
#define NQ      (B_ * M_PER)        // 16384 total queries

// d_out layout: x_out | pos_out | batch_out (floats)
#define POS_OFF   ((size_t)NQ * F_OUT)
#define BATCH_OFF (POS_OFF + (size_t)NQ * 3)

typedef __attribute__((ext_vector_type(2))) float v2f;
typedef __attribute__((ext_vector_type(8))) float v8f;

// D(16x16,f32) += A(16x4,f32) x B(4x16,f32)  -- CDNA5 V_WMMA_F32_16X16X4_F32
__device__ __forceinline__ v8f wmma4(v2f a, v2f b, v8f c) {
  return __builtin_amdgcn_wmma_f32_16x16x4_f32(false, a, false, b, (short)0, c,
                                               false, false);
}

// ---------------------------------------------------------------------------
// Kernel 1: brute-force KNN (top-16 smallest d2) per query.
// 256 threads/block, 1 query/thread; position tiles staged through LDS.
// Also emits pos_out and batch_out.
// ---------------------------------------------------------------------------
#define KNN_TILE 256

__global__ __launch_bounds__(256) void knn_kernel(const float* __restrict__ pos,
                                                  int* __restrict__ nbr,
                                                  float* __restrict__ out) {
  __shared__ float px[KNN_TILE], py[KNN_TILE], pz[KNN_TILE];

  const int tid = threadIdx.x;
  const int q   = blockIdx.x * 256 + tid;     // all queries in a block share b
  const int b   = q >> 11;                    // q / M_PER
  const int m   = q & (M_PER - 1);
  const size_t bbase = (size_t)b * N_PER;
  const size_t qi    = bbase + (size_t)m * DEC;

  const float qx = pos[qi * 3 + 0];
  const float qy = pos[qi * 3 + 1];
  const float qz = pos[qi * 3 + 2];

  // Register-resident top-16, sorted descending (dist[0] == current max).
  float dist[KNBR];
  int   idx[KNBR];
#pragma unroll
  for (int s = 0; s < KNBR; ++s) { dist[s] = 3.402823e38f; idx[s] = 0; }

  for (int t0 = 0; t0 < N_PER; t0 += KNN_TILE) {
    __syncthreads();
    const size_t j = bbase + (size_t)(t0 + tid);
    px[tid] = pos[j * 3 + 0];
    py[tid] = pos[j * 3 + 1];
    pz[tid] = pos[j * 3 + 2];
    if (t0 + KNN_TILE < N_PER) {  // hint next tile into cache
      __builtin_prefetch(&pos[(bbase + (size_t)(t0 + KNN_TILE + tid)) * 3], 0, 1);
    }
    __syncthreads();

#pragma unroll 4
    for (int i = 0; i < KNN_TILE; ++i) {
      const float dx = qx - px[i];
      const float dy = qy - py[i];
      const float dz = qz - pz[i];
      const float d2 = dx * dx + dy * dy + dz * dz;
      if (d2 < dist[0]) {                 // rare path (~K*ln(N/K) times)
        dist[0] = d2;
        idx[0]  = t0 + i;
#pragma unroll
        for (int s = 0; s < KNBR - 1; ++s) {   // one bubble pass restores order
          if (dist[s] < dist[s + 1]) {
            float td = dist[s]; dist[s] = dist[s + 1]; dist[s + 1] = td;
            int   ti = idx[s];  idx[s]  = idx[s + 1];  idx[s + 1]  = ti;
          }
        }
      }
    }
  }

#pragma unroll
  for (int s = 0; s < KNBR; ++s) nbr[(size_t)q * KNBR + s] = idx[s];

  float* posOut = out + POS_OFF;
  posOut[(size_t)q * 3 + 0] = qx;
  posOut[(size_t)q * 3 + 1] = qy;
  posOut[(size_t)q * 3 + 2] = qz;
  out[BATCH_OFF + q] = (float)b;
}

// ---------------------------------------------------------------------------
// Kernel 2: per-query 2-layer MLP on the 16 gathered neighbors + max-pool.
// One wave32 per query; 8 waves (256 threads) per block.
//   Layer 1: (16x68) @ (68x64)   -> 4 N-tiles x 17 K-chunks of WMMA f32 16x16x4
//   Layer 2: (16x64) @ (64x128)  -> 8 N-tiles x 16 K-chunks
// A-operand layout (ISA 7.12.2, 32-bit A 16x4): lanes 0-15 hold K=k,k+1;
// lanes 16-31 hold K=k+2,k+3 (row M = lane%16) -> one float2 load per chunk.
// All control flow around the WMMAs is branch-free (EXEC stays all-ones).
// ---------------------------------------------------------------------------
#define QW 8                         // queries (waves) per block
#define HSTR 68                      // LDS stride: bank = 4*row+c, conflict-free

__global__ __launch_bounds__(256) void mlp_kernel(
    const float* __restrict__ x, const float* __restrict__ pos,
    const int* __restrict__ nbr,
    const float* __restrict__ W1, const float* __restrict__ b1,
    const float* __restrict__ W2, const float* __restrict__ b2,
    float* __restrict__ out) {
  __shared__ float hbuf[QW * 16 * HSTR];   // per-wave h transpose buffer (34.8 KB)

  const int tid  = threadIdx.x;
  const int lane = tid & 31;
  const int wv   = tid >> 5;
  const int row  = lane & 15;              // M index (neighbor slot)
  const int half = lane >> 4;              // 0: K=k,k+1   1: K=k+2,k+3

  const int q = blockIdx.x * QW + wv;
  const int b = q >> 11;
  const int m = q & (M_PER - 1);
  const size_t bbase = (size_t)b * N_PER;

  // This lane's neighbor row and relative position.
  const int    nk = nbr[(size_t)q * KNBR + row];
  const size_t qi = bbase + (size_t)m * DEC;
  const size_t ni = bbase + (size_t)nk;
  const float rx = pos[ni * 3 + 0] - pos[qi * 3 + 0];
  const float ry = pos[ni * 3 + 1] - pos[qi * 3 + 1];
  const float rz = pos[ni * 3 + 2] - pos[qi * 3 + 2];
  const float* __restrict__ xrow = x + ni * F_IN;

  // ---------------- Layer 1: msg(16x67) @ W1(67x64) ----------------
  v8f acc[4];
#pragma unroll
  for (int t = 0; t < 4; ++t) acc[t] = (v8f){0.f,0.f,0.f,0.f,0.f,0.f,0.f,0.f};

  for (int kc = 0; kc < F_IN; kc += 4) {                 // 16 feature chunks
    const v2f a = *(const v2f*)(xrow + kc + 2 * half);   // 8B aligned gather
#pragma unroll
    for (int t = 0; t < 4; ++t) {
      const int n  = t * 16 + row;
      const int kr = kc + 2 * half;
      v2f bv;
      bv.x = W1[(size_t)kr * HID + n];
      bv.y = W1[(size_t)(kr + 1) * HID + n];
      acc[t] = wmma4(a, bv, acc[t]);
    }
  }
  {  // tail chunk (cols 64..67): rel xyz + zero pad, fully branch-free.
    v2f a;
    a.x = half ? rz : rx;                                 // col 64 / 66
    a.y = half ? 0.f : ry;                                // col 65 / 67 (pad)
    const int kr = F_IN + 2 * half;                       // 64 or 66: both valid
#pragma unroll
    for (int t = 0; t < 4; ++t) {
      const int n = t * 16 + row;
      v2f bv;
      bv.x = W1[(size_t)kr * HID + n];
      const float w65 = W1[(size_t)65 * HID + n];         // valid for all lanes
      bv.y = half ? 0.f : w65;                            // row 67 is padding
      acc[t] = wmma4(a, bv, acc[t]);
    }
  }

  // bias + relu, transpose D (row-striped across lanes) into LDS A-layout.
  float* hb = hbuf + wv * 16 * HSTR;
#pragma unroll
  for (int t = 0; t < 4; ++t) {
    const float bias = b1[t * 16 + row];
#pragma unroll
    for (int e = 0; e < 8; ++e) {
      const float v = fmaxf(acc[t][e] + bias, 0.f);
      hb[(e + 8 * half) * HSTR + t * 16 + row] = v;       // h[M][col]
    }
  }
  __syncthreads();

  // ---------------- Layer 2: h(16x64) @ W2(64x128) ----------------
  v8f acc2[8];
#pragma unroll
  for (int t = 0; t < 8; ++t) acc2[t] = (v8f){0.f,0.f,0.f,0.f,0.f,0.f,0.f,0.f};

  for (int kc = 0; kc < HID; kc += 4) {                   // 16 chunks
    const v2f a = *(const v2f*)(hb + row * HSTR + kc + 2 * half);
#pragma unroll
    for (int t = 0; t < 8; ++t) {
      const int n  = t * 16 + row;
      const int kr = kc + 2 * half;
      v2f bv;
      bv.x = W2[(size_t)kr * F_OUT + n];
      bv.y = W2[(size_t)(kr + 1) * F_OUT + n];
      acc2[t] = wmma4(a, bv, acc2[t]);
    }
  }

  // bias + relu + max over the 16 neighbor rows.
  // Lane L holds rows 0..7 (its 8 acc elems), lane L^16 holds rows 8..15 of
  // the same column -> local fold then cross-half shuffle (wave32).
#pragma unroll
  for (int t = 0; t < 8; ++t) {
    const float bias = b2[t * 16 + row];
    float mv = 0.f;                                       // relu floor
#pragma unroll
    for (int e = 0; e < 8; ++e) mv = fmaxf(mv, acc2[t][e] + bias);
    const float other = __shfl_xor(mv, 16, 32);
    mv = fmaxf(mv, other);
    if (half == 0) out[(size_t)q * F_OUT + t * 16 + row] = mv;
  }
}

// ---------------------------------------------------------------------------
// Host launcher
// ---------------------------------------------------------------------------
extern "C" void kernel_launch(void* const* d_in, const int* in_sizes, int n_in,
                              void* d_out, int out_size, void* d_ws, size_t ws_size,
                              hipStream_t stream) {
  (void)in_sizes; (void)n_in; (void)out_size; (void)ws_size;
  const float* x   = (const float*)d_in[0];
  const float* pos = (const float*)d_in[1];
  // d_in[2] = batch indices: unused (deterministic layout), like the reference.
  const float* W1  = (const float*)d_in[3];
  const float* b1  = (const float*)d_in[4];
  const float* W2  = (const float*)d_in[5];
  const float* b2  = (const float*)d_in[6];
  float* out = (float*)d_out;
  int*   nbr = (int*)d_ws;                 // NQ*KNBR ints = 1 MB scratch

  knn_kernel<<<NQ / 256, 256, 0, stream>>>(pos, nbr, out);
  mlp_kernel<<<NQ / QW, 256, 0, stream>>>(x, pos, nbr, W1, b1, W2, b2, out);
}